// GINEBackbone_11716670783506
// MI455X (gfx1250) — compile-verified
//
#include <hip/hip_runtime.h>

// Problem constants from the reference
#define N_NODES 50000
#define N_EDGES 1600000
#define DIM     128
#define LAYERS  3
#define EPS_BN  1e-5f

typedef __attribute__((ext_vector_type(2))) float v2f;
typedef __attribute__((ext_vector_type(8))) float v8f;

// ---------------------------------------------------------------------------
// agg[i] = (1 + eps[l]) * h[i]   (folds the GIN self-term into the scatter buf)
// ---------------------------------------------------------------------------
__global__ __launch_bounds__(256)
void init_agg_kernel(const float* __restrict__ h, const float* __restrict__ eps,
                     int l, float* __restrict__ agg, int total4) {
    int i = blockIdx.x * blockDim.x + threadIdx.x;
    if (i >= total4) return;
    float s = 1.0f + eps[l];
    float4 v = ((const float4*)h)[i];
    v.x *= s; v.y *= s; v.z *= s; v.w *= s;
    ((float4*)agg)[i] = v;
}

__global__ void zero_stats_kernel(float* __restrict__ stats) {
    stats[threadIdx.x] = 0.0f;
}

// ---------------------------------------------------------------------------
// One wave32 per edge; lane handles 4 consecutive features.
// msg = relu(h[src] + ea[e]*we + be)  scattered into agg[dst] with f32 atomics.
// h (25.6MB) and agg are L2-resident on MI455X (192MB L2).
// ---------------------------------------------------------------------------
__global__ __launch_bounds__(256)
void edge_kernel(const float* __restrict__ h, const float* __restrict__ ea,
                 const int* __restrict__ src, const int* __restrict__ dst,
                 const float* __restrict__ we, const float* __restrict__ be,
                 float* __restrict__ agg, int E) {
    int tid  = blockIdx.x * blockDim.x + threadIdx.x;
    int e    = tid >> 5;
    int lane = tid & 31;
    if (e >= E) return;

    int s = src[e];
    int d = dst[e];
    float a = ea[e];
    int f0 = lane * 4;

    float4 hv = *(const float4*)(h  + (size_t)s * DIM + f0);
    float4 w4 = *(const float4*)(we + f0);
    float4 b4 = *(const float4*)(be + f0);

    float4 m;
    m.x = fmaxf(hv.x + a * w4.x + b4.x, 0.0f);
    m.y = fmaxf(hv.y + a * w4.y + b4.y, 0.0f);
    m.z = fmaxf(hv.z + a * w4.z + b4.z, 0.0f);
    m.w = fmaxf(hv.w + a * w4.w + b4.w, 0.0f);

    float* outp = agg + (size_t)d * DIM + f0;
    atomicAdd(outp + 0, m.x);
    atomicAdd(outp + 1, m.y);
    atomicAdd(outp + 2, m.z);
    atomicAdd(outp + 3, m.w);
}

// ---------------------------------------------------------------------------
// out = [relu](Z @ W^T + bias), Z:(N,128), W:(128,128) row-major.
// Block = 16 rows of Z staged in LDS via GLOBAL_LOAD_ASYNC_TO_LDS_B128
// (memory->LDS direct, ASYNCcnt-tracked, no VGPR round trip). The lz pointer
// is passed into the asm so the LDS buffer escapes and the compiler cannot
// fold the subsequent ds_loads of the A tile to undef.
// Pitch 132 -> conflict-free 16-lane column reads.
// 8 waves x one 16x16 tile each (full 128 output cols).
// Full fp32 precision via V_WMMA_F32_16X16X4_F32, K-loop of 32 steps.
// EXEC is all-ones: N is an exact multiple of 16, no divergence in this kernel.
// ---------------------------------------------------------------------------
#define LZ_PITCH 132

__global__ __launch_bounds__(256)
void gemm_wmma_kernel(const float* __restrict__ Z, const float* __restrict__ W,
                      const float* __restrict__ bias, float* __restrict__ out,
                      int do_relu) {
    __shared__ float lz[16 * LZ_PITCH];   // only LDS object -> base offset 0

    int tid  = threadIdx.x;
    int row0 = blockIdx.x * 16;

    // Async-stage the 16x128 Z tile: 512 float4 slots, 2 per thread.
    // GVS mode: mem_addr = SGPR_base + VGPR_offset ; LDS dest = per-lane VDST.
    // Operand %3 (flat lz pointer) is unused in the template but forces the
    // LDS array to escape so the "memory" clobber covers it.
    #pragma unroll
    for (int i = 0; i < 2; ++i) {
        int lin = tid + 256 * i;        // float4 index
        int r   = lin >> 5;             // 32 float4 per row
        int c   = (lin & 31) << 2;
        unsigned lds_off = (unsigned)((r * LZ_PITCH + c) * 4);
        unsigned voff    = (unsigned)(((row0 + r) * DIM + c) * 4);
        asm volatile("global_load_async_to_lds_b128 %0, %1, %2"
                     :: "v"(lds_off), "v"(voff), "s"(Z), "v"(lz)
                     : "memory");
    }
    // ASYNCcnt is NOT covered by the barrier's dscnt wait — drain it explicitly.
    asm volatile("s_wait_asynccnt 0x0" ::: "memory");
    __syncthreads();

    int wave = tid >> 5;
    int lane = tid & 31;
    int hi   = lane >> 4;       // 0: lanes 0-15, 1: lanes 16-31
    int m    = lane & 15;       // A-row / B-col / output col within tile
    int n0   = wave * 16;       // this wave's output-column tile

    // out[m][n] = sum_k Z[m][k] * W[n][k]  ->  B[k][n] = W[n][k]
    const float* wrow = W + (size_t)(n0 + m) * DIM;

    v8f acc = {0.f, 0.f, 0.f, 0.f, 0.f, 0.f, 0.f, 0.f};

    #pragma unroll 4
    for (int k0 = 0; k0 < DIM; k0 += 4) {
        int k = k0 + 2 * hi;
        // A 16x4 f32 layout (ISA 7.12.2): lane m, VGPR{0,1} = K = k0+2*hi+{0,1}
        v2f a = *(const v2f*)(&lz[m * LZ_PITCH + k]);
        // B 4x16 f32, mirrored striping
        v2f b = *(const v2f*)(wrow + k);
        __builtin_prefetch(wrow + k + 8, 0, 3);   // prefetch next K block toward near caches
        acc = __builtin_amdgcn_wmma_f32_16x16x4_f32(
            /*neg_a=*/false, a, /*neg_b=*/false, b,
            /*c_mod=*/(short)0, acc, /*reuse_a=*/false, /*reuse_b=*/false);
    }

    // C/D layout: VGPR r -> M = r + 8*hi, N = lane&15
    float bn = bias[n0 + m];
    #pragma unroll
    for (int r = 0; r < 8; ++r) {
        int M = r + 8 * hi;
        float v = acc[r] + bn;
        if (do_relu) v = fmaxf(v, 0.0f);
        out[(size_t)(row0 + M) * DIM + n0 + m] = v;
    }
}

// ---------------------------------------------------------------------------
// Per-feature sum / sum-of-squares reduction (coalesced: thread d -> feature d)
// ---------------------------------------------------------------------------
__global__ __launch_bounds__(256)
void bn_stats_kernel(const float* __restrict__ z, float* __restrict__ stats) {
    int d       = threadIdx.x & (DIM - 1);
    int rowlane = threadIdx.x >> 7;                 // 0 or 1
    int stride  = gridDim.x * 2;
    float s = 0.0f, s2 = 0.0f;
    for (int r = blockIdx.x * 2 + rowlane; r < N_NODES; r += stride) {
        float v = z[(size_t)r * DIM + d];
        s  += v;
        s2 += v * v;
    }
    atomicAdd(&stats[d], s);
    atomicAdd(&stats[DIM + d], s2);
}

// ---------------------------------------------------------------------------
// h = relu((z - mu) * rsqrt(var + eps) * gamma + beta), written straight into
// this layer's d_out slice (which is also the next layer's input).
// ---------------------------------------------------------------------------
__global__ __launch_bounds__(256)
void bn_apply_kernel(const float* __restrict__ z, const float* __restrict__ stats,
                     const float* __restrict__ gamma, const float* __restrict__ beta,
                     float* __restrict__ out, int total) {
    int i = blockIdx.x * blockDim.x + threadIdx.x;
    if (i >= total) return;
    int d = i & (DIM - 1);
    const float inv_n = 1.0f / (float)N_NODES;
    float mu  = stats[d] * inv_n;
    float var = stats[DIM + d] * inv_n - mu * mu;
    float v = (z[i] - mu) * rsqrtf(var + EPS_BN) * gamma[d] + beta[d];
    out[i] = fmaxf(v, 0.0f);
}

// ---------------------------------------------------------------------------
extern "C" void kernel_launch(void* const* d_in, const int* in_sizes, int n_in,
                              void* d_out, int out_size, void* d_ws, size_t ws_size,
                              hipStream_t stream) {
    const float* x       = (const float*)d_in[0];
    const float* ea      = (const float*)d_in[1];
    const float* lin_e_w = (const float*)d_in[2];   // (L, D, 1)
    const float* lin_e_b = (const float*)d_in[3];   // (L, D)
    const float* eps     = (const float*)d_in[4];   // (L,)
    const float* W1      = (const float*)d_in[5];   // (L, D, D)
    const float* b1      = (const float*)d_in[6];   // (L, D)
    const float* W2      = (const float*)d_in[7];   // (L, D, D)
    const float* b2      = (const float*)d_in[8];   // (L, D)
    const float* gamma   = (const float*)d_in[9];   // (L, D)
    const float* beta    = (const float*)d_in[10];  // (L, D)
    const int*   eidx    = (const int*)d_in[11];    // (2, E)

    const int* src = eidx;
    const int* dst = eidx + N_EDGES;

    float* out = (float*)d_out;

    // Workspace: two N*D fp32 buffers (ping-pong) + 256-float BN stats
    float* A     = (float*)d_ws;                    // agg / z2 buffer
    float* z1    = A  + (size_t)N_NODES * DIM;      // post-GEMM1 buffer
    float* stats = z1 + (size_t)N_NODES * DIM;      // [sum(128), sumsq(128)]

    const int ND = N_NODES * DIM;
    const int init_blocks  = (ND / 4 + 255) / 256;
    const int edge_blocks  = (int)(((size_t)N_EDGES * 32 + 255) / 256);
    const int gemm_blocks  = N_NODES / 16;          // 3125, exact
    const int apply_blocks = (ND + 255) / 256;

    const float* h = x;
    for (int l = 0; l < LAYERS; ++l) {
        const float* we_l = lin_e_w + (size_t)l * DIM;
        const float* be_l = lin_e_b + (size_t)l * DIM;
        const float* W1_l = W1 + (size_t)l * DIM * DIM;
        const float* b1_l = b1 + (size_t)l * DIM;
        const float* W2_l = W2 + (size_t)l * DIM * DIM;
        const float* b2_l = b2 + (size_t)l * DIM;
        const float* g_l  = gamma + (size_t)l * DIM;
        const float* bt_l = beta  + (size_t)l * DIM;
        float* hl = out + (size_t)l * ND;

        // agg = (1+eps)*h ; stats = 0
        init_agg_kernel<<<init_blocks, 256, 0, stream>>>(h, eps, l, A, ND / 4);
        zero_stats_kernel<<<1, 256, 0, stream>>>(stats);

        // agg += scatter_sum(relu(h[src] + ea*we + be), dst)
        edge_kernel<<<edge_blocks, 256, 0, stream>>>(h, ea, src, dst, we_l, be_l, A, N_EDGES);

        // z1 = relu(agg @ W1^T + b1)   (fp32 WMMA)
        gemm_wmma_kernel<<<gemm_blocks, 256, 0, stream>>>(A, W1_l, b1_l, z1, 1);
        // A  = z1 @ W2^T + b2          (fp32 WMMA, reuse agg buffer)
        gemm_wmma_kernel<<<gemm_blocks, 256, 0, stream>>>(z1, W2_l, b2_l, A, 0);

        // BatchNorm + relu -> layer output slice (= next layer input)
        bn_stats_kernel<<<256, 256, 0, stream>>>(A, stats);
        bn_apply_kernel<<<apply_blocks, 256, 0, stream>>>(A, stats, g_l, bt_l, hl, ND);

        h = hl;
    }
}